// GIN_395136991277
// MI455X (gfx1250) — compile-verified
//
#include <hip/hip_runtime.h>

typedef __attribute__((ext_vector_type(2))) float v2f;
typedef __attribute__((ext_vector_type(8))) float v8f;

#define N_NODES  50000
#define N_EDGES  800000
#define N_GRAPHS 512
#define BN_EPS   1e-5f

// ---------------------------------------------------------------- utilities
__global__ void zero_f32(float* __restrict__ p, int n4) {
    int t = blockIdx.x * blockDim.x + threadIdx.x;
    if (t < n4) ((float4*)p)[t] = make_float4(0.f, 0.f, 0.f, 0.f);
}

// x[node, 0:128] = emb[ids[node], 0:128]; 32 lanes * float4 per node
__global__ void gather_emb(const int* __restrict__ ids, const float* __restrict__ emb,
                           float* __restrict__ x) {
    int t = blockIdx.x * blockDim.x + threadIdx.x;
    int node = t >> 5, lane = t & 31;
    if (node >= N_NODES) return;
    int id = ids[node];
    float4 v = *(const float4*)(emb + (long)id * 128 + lane * 4);
    *(float4*)(x + (long)node * 128 + lane * 4) = v;
}

// one wave per edge: agg[dst] += x[src]  (128 floats, float4 per lane)
__global__ void edge_scatter(const float* __restrict__ x, const int* __restrict__ ei,
                             float* __restrict__ agg) {
    int wave = threadIdx.x >> 5, lane = threadIdx.x & 31;
    int e = blockIdx.x * 8 + wave;
    if (e >= N_EDGES) return;
    int s = ei[e];
    int d = ei[N_EDGES + e];
    float4 v = *(const float4*)(x + (long)s * 128 + lane * 4);
    float* p = agg + (long)d * 128 + lane * 4;
    atomicAdd(p + 0, v.x);
    atomicAdd(p + 1, v.y);
    atomicAdd(p + 2, v.z);
    atomicAdd(p + 3, v.w);
}

// ---------------------------------------------------------------- WMMA GEMM
// O = epilogue( (X [+ AGG]) @ W + bias ), one wave32 per 16x16 tile.
// M, N multiples of 16; K multiple of 4. Exact fp32 via V_WMMA_F32_16X16X4_F32.
// A frag (16x4 f32): lanes 0-15 -> M=0..15, VGPR{0,1}=K{0,1}; lanes 16-31 -> K{2,3}.
// C/D (16x16 f32): lane half selects M base (0 / 8), VGPR r -> M=r(+8), lane&15 -> N.
template <bool FUSE_ADD, bool BN, bool RELU>
__global__ void gemm_wmma(const float* __restrict__ X, const float* __restrict__ AGG,
                          const float* __restrict__ W, const float* __restrict__ bias,
                          const float* __restrict__ gamma, const float* __restrict__ beta,
                          const float* __restrict__ mean, const float* __restrict__ var,
                          float* __restrict__ O, int M, int K, int N) {
    int wave = threadIdx.x >> 5, lane = threadIdx.x & 31;
    int tilesN = N >> 4;
    int tile = blockIdx.x * 8 + wave;
    if (tile >= (M >> 4) * tilesN) return;           // whole-wave exit: EXEC stays all-ones
    int mt = tile / tilesN, nt = tile % tilesN;
    int mi = lane & 15, kh = lane >> 4;              // mi doubles as B's column index

    const float* xr = X + (long)(mt * 16 + mi) * K + kh * 2;
    const float* ar = FUSE_ADD ? (AGG + (long)(mt * 16 + mi) * K + kh * 2) : nullptr;
    const float* wc = W + (long)(kh * 2) * N + nt * 16 + mi;

    v8f acc = {0.f, 0.f, 0.f, 0.f, 0.f, 0.f, 0.f, 0.f};
    for (int kb = 0; kb < K; kb += 4) {
        v2f a, b;
        a.x = xr[kb];
        a.y = xr[kb + 1];
        if (FUSE_ADD) { a.x += ar[kb]; a.y += ar[kb + 1]; }
        b.x = wc[(long)kb * N];
        b.y = wc[(long)(kb + 1) * N];
        acc = __builtin_amdgcn_wmma_f32_16x16x4_f32(false, a, false, b,
                                                    (short)0, acc, false, false);
    }

    int col = nt * 16 + mi;
    float bb = bias[col];
    float mu = 0.f, inv = 1.f, gm = 1.f, bt = 0.f;
    if (BN) { mu = mean[col]; inv = rsqrtf(var[col] + BN_EPS); gm = gamma[col]; bt = beta[col]; }
    float* orow = O + (long)(mt * 16 + kh * 8) * N + col;
#pragma unroll
    for (int r = 0; r < 8; r++) {
        float v = acc[r] + bb;
        if (BN)   v = (v - mu) * inv * gm + bt;
        if (RELU) v = fmaxf(v, 0.f);
        orow[(long)r * N] = v;
    }
}

// ---------------------------------------------------------------- pooling + head
// gcat[batch[n], layer*128 + c] += h_layer[n, c]  (batch sorted, atomics)
__global__ void pool3(const float* __restrict__ h1, const float* __restrict__ h2,
                      const float* __restrict__ h3, const int* __restrict__ batch,
                      float* __restrict__ gcat) {
    int t = blockIdx.x * blockDim.x + threadIdx.x;
    int node = t >> 5, lane = t & 31;
    if (node >= N_NODES) return;
    int g = batch[node];
    float* base = gcat + (long)g * 384 + lane * 4;
    const float* hs[3] = {h1, h2, h3};
#pragma unroll
    for (int l = 0; l < 3; l++) {
        float4 v = *(const float4*)(hs[l] + (long)node * 128 + lane * 4);
        float* p = base + l * 128;
        atomicAdd(p + 0, v.x);
        atomicAdd(p + 1, v.y);
        atomicAdd(p + 2, v.z);
        atomicAdd(p + 3, v.w);
    }
}

// out[g, o] = h[g, :384] . w[:, o] + b[o]   (N=10, not WMMA-shaped)
__global__ void lin2_kernel(const float* __restrict__ h, const float* __restrict__ w,
                            const float* __restrict__ b, float* __restrict__ out) {
    int t = blockIdx.x * blockDim.x + threadIdx.x;
    if (t >= N_GRAPHS * 10) return;
    int g = t / 10, o = t % 10;
    const float* hr = h + (long)g * 384;
    float acc = b[o];
    for (int k = 0; k < 384; k++) acc = fmaf(hr[k], w[k * 10 + o], acc);
    out[t] = acc;
}

// ---------------------------------------------------------------- launcher
extern "C" void kernel_launch(void* const* d_in, const int* in_sizes, int n_in,
                              void* d_out, int out_size, void* d_ws, size_t ws_size,
                              hipStream_t stream) {
    const int*   x_ids = (const int*)d_in[0];
    const int*   edge  = (const int*)d_in[1];
    const int*   batch = (const int*)d_in[2];
    const float* emb   = (const float*)d_in[3];

    const float *c_w1[3], *c_b1[3], *c_g[3], *c_be[3], *c_mu[3], *c_va[3], *c_w2[3], *c_b2[3];
    for (int i = 0; i < 3; i++) {
        c_w1[i] = (const float*)d_in[4 + 8 * i + 0];
        c_b1[i] = (const float*)d_in[4 + 8 * i + 1];
        c_g[i]  = (const float*)d_in[4 + 8 * i + 2];
        c_be[i] = (const float*)d_in[4 + 8 * i + 3];
        c_mu[i] = (const float*)d_in[4 + 8 * i + 4];
        c_va[i] = (const float*)d_in[4 + 8 * i + 5];
        c_w2[i] = (const float*)d_in[4 + 8 * i + 6];
        c_b2[i] = (const float*)d_in[4 + 8 * i + 7];
    }
    const float* lin1_w = (const float*)d_in[28];
    const float* lin1_b = (const float*)d_in[29];
    const float* lin2_w = (const float*)d_in[30];
    const float* lin2_b = (const float*)d_in[31];

    // workspace layout: 5 node-sized fp32 buffers (buf0 is x0, recycled for h3)
    const long NF = (long)N_NODES * 128;            // 6.4M floats each
    float* ws   = (float*)d_ws;
    float* x0   = ws + 0 * NF;                       // x0, later h3
    float* agg  = ws + 1 * NF;                       // agg, later gcat
    float* tmp  = ws + 2 * NF;                       // mlp hidden, later lin1 out
    float* hbuf[3] = { ws + 3 * NF, ws + 4 * NF, x0 };
    float* gcat  = agg;
    float* lin1o = tmp;

    gather_emb<<<(N_NODES * 32 + 255) / 256, 256, 0, stream>>>(x_ids, emb, x0);

    const int tiles_node = (N_NODES / 16) * (128 / 16);   // 25000 tiles, 8 waves/block
    const float* xin = x0;
    for (int l = 0; l < 3; l++) {
        zero_f32<<<(int)((NF / 4 + 255) / 256), 256, 0, stream>>>(agg, (int)(NF / 4));
        edge_scatter<<<N_EDGES / 8, 256, 0, stream>>>(xin, edge, agg);
        gemm_wmma<true, true, true><<<(tiles_node + 7) / 8, 256, 0, stream>>>(
            xin, agg, c_w1[l], c_b1[l], c_g[l], c_be[l], c_mu[l], c_va[l],
            tmp, N_NODES, 128, 128);
        gemm_wmma<false, false, true><<<(tiles_node + 7) / 8, 256, 0, stream>>>(
            tmp, nullptr, c_w2[l], c_b2[l], nullptr, nullptr, nullptr, nullptr,
            hbuf[l], N_NODES, 128, 128);
        xin = hbuf[l];
    }

    zero_f32<<<(N_GRAPHS * 384 / 4 + 255) / 256, 256, 0, stream>>>(gcat, N_GRAPHS * 384 / 4);
    pool3<<<(N_NODES * 32 + 255) / 256, 256, 0, stream>>>(hbuf[0], hbuf[1], hbuf[2], batch, gcat);

    const int tiles_head = (N_GRAPHS / 16) * (384 / 16);  // 768 tiles
    gemm_wmma<false, false, true><<<(tiles_head + 7) / 8, 256, 0, stream>>>(
        gcat, nullptr, lin1_w, lin1_b, nullptr, nullptr, nullptr, nullptr,
        lin1o, N_GRAPHS, 384, 384);

    lin2_kernel<<<(N_GRAPHS * 10 + 255) / 256, 256, 0, stream>>>(lin1o, lin2_w, lin2_b,
                                                                 (float*)d_out);
}